// RegionProposalNetwork_10952166605182
// MI455X (gfx1250) — compile-verified
//
#include <hip/hip_runtime.h>
#include <math.h>
#include <stdint.h>

// ---------------------------------------------------------------------------
// RPN forward (torchvision semantics) for MI455X / gfx1250.
//
// Pipeline (all on `stream`):
//   1) init_ws:            zero counters/hists, poison candidate slots
//   2) 4x (hist_pass ->    8-bit radix-select of the 1000th-largest raw
//         pick_digit)      objectness key per (image, level)  [10 pairs]
//   3) compact_decode:     emit exactly 1000 decoded boxes + raw scores
//                          per (image, level) into ws (set, order-free)
//   4) nms_kernel:         1 block/image; async-DMA candidate boxes into LDS,
//                          sigmoid/clip/filter, 1000 sequential greedy-NMS
//                          iterations with wave32 shfl argmax, write [1000,5]
//
// Why no WMMA: zero matmul FLOPs in this op. The gfx1250-specific paths used
// are GLOBAL_LOAD_ASYNC_TO_LDS_B128 + S_WAIT_ASYNCCNT and 320KB-class LDS.
// ---------------------------------------------------------------------------

#define NIMG      2
#define NLVL      5
#define KPER      1000
#define KCAND     (NLVL * KPER)        // 5000 candidates / image
#define POSTN     1000
#define IMG_DIM   2048.0f
#define NMS_TH    0.7f
#define XFORM_CLIP 4.135166556742356f  // log(1000/16)

struct Ptrs {
    const float* obj[NLVL];
    const float* box[NLVL];
};

// level lvl: grid W=H=512>>lvl, stride=4<<lvl, size=32<<lvl, A=3 aspects

__device__ __forceinline__ unsigned f2key(float f) {
    unsigned u = __float_as_uint(f);
    return (u & 0x80000000u) ? ~u : (u | 0x80000000u);   // monotone: bigger float -> bigger key
}

// ---------------------------------------------------------------- init ----
__global__ void init_ws_kernel(float* candBox, float* candScore, unsigned* hist,
                               unsigned* prefix, unsigned* krem,
                               unsigned* cntgt, unsigned* cnteq) {
    int t = blockIdx.x * blockDim.x + threadIdx.x;
    if (t < NIMG * KCAND * 4) candBox[t] = 0.0f;
    if (t < NIMG * KCAND)     candScore[t] = -__builtin_inff();
    if (t < NIMG * NLVL * 256) hist[t] = 0u;
    if (t < NIMG * NLVL) { prefix[t] = 0u; krem[t] = KPER; cntgt[t] = 0u; cnteq[t] = 0u; }
}

// ------------------------------------------------------------ histogram ----
__global__ void hist_pass_kernel(Ptrs p, int pass, unsigned* hist, const unsigned* prefix) {
    __shared__ unsigned h[256];
    const int pair = blockIdx.y;                 // n*NLVL + lvl? -> use n = pair/NLVL
    const int n = pair / NLVL, lvl = pair % NLVL;
    const int W = 512 >> lvl, HW = W * W, cnt = 3 * HW;
    const float* __restrict__ obj = p.obj[lvl];

    h[threadIdx.x] = 0u;
    __syncthreads();

    const unsigned pref = prefix[pair];
    const int shift = 24 - 8 * pass;
    for (int i = blockIdx.x * blockDim.x + threadIdx.x; i < cnt; i += gridDim.x * blockDim.x) {
        int a = i % 3, hw = i / 3;
        unsigned key = f2key(obj[(size_t)(n * 3 + a) * HW + hw]);
        bool ok = (pass == 0) || ((key >> (shift + 8)) == pref);
        if (ok) atomicAdd(&h[(key >> shift) & 255u], 1u);
    }
    __syncthreads();
    if (h[threadIdx.x]) atomicAdd(&hist[pair * 256 + threadIdx.x], h[threadIdx.x]);
}

// ---------------------------------------------------------- digit pick ----
__global__ void pick_digit_kernel(unsigned* hist, unsigned* prefix, unsigned* krem) {
    const int pair = blockIdx.x;
    unsigned* h = hist + pair * 256;
    unsigned kr = krem[pair];
    unsigned cum = 0, above = 0;
    int chosen = -1;
    for (int dig = 255; dig >= 0; --dig) {
        unsigned c = h[dig];
        h[dig] = 0u;                       // reset for next pass
        if (chosen < 0) {
            if (cum + c >= kr) { chosen = dig; above = cum; }
            cum += c;
        }
    }
    if (chosen < 0) { chosen = 0; above = cum; }   // unreachable (cnt >= 3072 > 1000)
    prefix[pair] = (prefix[pair] << 8) | (unsigned)chosen;
    krem[pair]   = kr - above;             // after last pass: #needed from the ==T bin
}

// ------------------------------------------------- compaction + decode ----
__global__ void compact_decode_kernel(Ptrs p, const unsigned* prefix, const unsigned* krem,
                                      unsigned* cntgt, unsigned* cnteq,
                                      float* candBox, float* candScore) {
    const int pair = blockIdx.y;
    const int n = pair / NLVL, lvl = pair % NLVL;
    const int W = 512 >> lvl, HW = W * W, cnt = 3 * HW;
    const unsigned T = prefix[pair];                 // full 32-bit key of 1000th-largest
    const unsigned G = KPER - krem[pair];            // count strictly greater than T
    const float strd = (float)(4 << lvl);
    const float sz   = (float)(32 << lvl);
    const float* __restrict__ obj = p.obj[lvl];
    const float* __restrict__ bx  = p.box[lvl];

    for (int i = blockIdx.x * blockDim.x + threadIdx.x; i < cnt; i += gridDim.x * blockDim.x) {
        int a = i % 3, hw = i / 3;
        float s = obj[(size_t)(n * 3 + a) * HW + hw];
        unsigned key = f2key(s);
        int slot = -1;
        if (key > T) {
            slot = (int)atomicAdd(&cntgt[pair], 1u);                // < G guaranteed
        } else if (key == T) {
            unsigned e = atomicAdd(&cnteq[pair], 1u);
            if (G + e < (unsigned)KPER) slot = (int)(G + e);
        }
        if (slot >= 0) {
            int hh = hw / W, ww = hw % W;
            // anchor: round-half-even matches jnp.round; cx/cy are exact shifts
            float asp = (a == 0) ? 0.5f : (a == 1 ? 1.0f : 2.0f);
            float hr = sqrtf(asp);
            float wa = 2.0f * rintf(0.5f * (sz / hr));
            float ha = 2.0f * rintf(0.5f * (sz * hr));
            float cxa = (float)ww * strd, cya = (float)hh * strd;
            size_t b0 = (size_t)(n * 12 + a * 4) * HW + hw;
            float dx = bx[b0], dy = bx[b0 + HW];
            float dw = fminf(bx[b0 + 2 * (size_t)HW], XFORM_CLIP);
            float dh = fminf(bx[b0 + 3 * (size_t)HW], XFORM_CLIP);
            float cx = dx * wa + cxa, cy = dy * ha + cya;
            float w2 = 0.5f * expf(dw) * wa, h2 = 0.5f * expf(dh) * ha;
            int o = n * KCAND + lvl * KPER + slot;
            candBox[o * 4 + 0] = cx - w2;
            candBox[o * 4 + 1] = cy - h2;
            candBox[o * 4 + 2] = cx + w2;
            candBox[o * 4 + 3] = cy + h2;
            candScore[o] = s;
        }
    }
}

// ------------------------------------------------------------------ NMS ----
__global__ __launch_bounds__(1024, 1)
void nms_kernel(const float* __restrict__ gBox, const float* __restrict__ gScore,
                float* __restrict__ out) {
    __shared__ float4 sbox[KCAND];     // clipped boxes (80 KB)
    __shared__ float  sms[KCAND];      // running masked scores (20 KB)
    __shared__ float  ssig[KCAND];     // sigmoid scores for output (20 KB)
    __shared__ int    selIdx[POSTN];
    __shared__ float  selVal[POSTN];
    __shared__ float  redS[32];
    __shared__ int    redI[32];

    const int n = blockIdx.x;
    const int tid = threadIdx.x;
    const float* base = gBox + (size_t)n * KCAND * 4;

    // --- stage raw candidate boxes into LDS via CDNA5 async global->LDS DMA
    for (int i = tid; i < KCAND; i += 1024) {
        unsigned lds_addr = (unsigned)(uintptr_t)(&sbox[i]);   // AS(3) byte offset
        unsigned goff = (unsigned)(i * 16);
        asm volatile("global_load_async_to_lds_b128 %0, %1, %2 offset:0"
                     :: "v"(lds_addr), "v"(goff), "s"(base)
                     : "memory");
    }
    asm volatile("s_wait_asynccnt 0" ::: "memory");
    __syncthreads();

    // --- sigmoid / clip / validity (in place in LDS)
    for (int i = tid; i < KCAND; i += 1024) {
        float4 b = sbox[i];
        float x1 = fminf(fmaxf(b.x, 0.0f), IMG_DIM);
        float y1 = fminf(fmaxf(b.y, 0.0f), IMG_DIM);
        float x2 = fminf(fmaxf(b.z, 0.0f), IMG_DIM);
        float y2 = fminf(fmaxf(b.w, 0.0f), IMG_DIM);
        float raw = gScore[(size_t)n * KCAND + i];
        float sc = 1.0f / (1.0f + expf(-raw));     // raw=-inf slots -> sc=0 -> dropped
        bool keep = ((x2 - x1) >= 1e-3f) && ((y2 - y1) >= 1e-3f) && (sc > 0.0f);
        sbox[i] = make_float4(x1, y1, x2, y2);
        ssig[i] = sc;
        sms[i]  = keep ? sc : -__builtin_inff();
    }
    __syncthreads();

    // --- 1000 sequential greedy iterations (the latency-critical chain)
    const int wave = tid >> 5, lane = tid & 31;
    for (int it = 0; it < POSTN; ++it) {
        // local argmax (ties -> lowest index; strided i is increasing per thread)
        float bs = -__builtin_inff(); int bi = 0x7fffffff;
        for (int i = tid; i < KCAND; i += 1024) {
            float s = sms[i];
            if (s > bs || (s == bs && i < bi)) { bs = s; bi = i; }
        }
        // wave32 reduction
        #pragma unroll
        for (int off = 16; off > 0; off >>= 1) {
            float os = __shfl_xor(bs, off, 32);
            int   oi = __shfl_xor(bi, off, 32);
            if (os > bs || (os == bs && oi < bi)) { bs = os; bi = oi; }
        }
        if (lane == 0) { redS[wave] = bs; redI[wave] = bi; }
        __syncthreads();
        if (wave == 0) {
            bs = redS[lane]; bi = redI[lane];
            #pragma unroll
            for (int off = 16; off > 0; off >>= 1) {
                float os = __shfl_xor(bs, off, 32);
                int   oi = __shfl_xor(bi, off, 32);
                if (os > bs || (os == bs && oi < bi)) { bs = os; bi = oi; }
            }
            if (lane == 0) { redS[0] = bs; redI[0] = bi; }
        }
        __syncthreads();
        bs = redS[0]; bi = redI[0];
        if (tid == 0) { selIdx[it] = bi; selVal[it] = (bs > -__builtin_inff()) ? 1.0f : 0.0f; }

        // suppression: level offsets (4097*lvl) guarantee zero cross-level IoU,
        // and cancel within a level -> only same-level IoU on clipped boxes.
        float4 bb = sbox[bi];
        float areaB = fmaxf(bb.z - bb.x, 0.0f) * fmaxf(bb.w - bb.y, 0.0f);
        int lvlB = bi / KPER;
        for (int i = tid; i < KCAND; i += 1024) {
            if (i / KPER == lvlB) {
                float4 c = sbox[i];
                float iw = fminf(c.z, bb.z) - fmaxf(c.x, bb.x);
                float ih = fminf(c.w, bb.w) - fmaxf(c.y, bb.y);
                float inter = fmaxf(iw, 0.0f) * fmaxf(ih, 0.0f);
                float areaI = fmaxf(c.z - c.x, 0.0f) * fmaxf(c.w - c.y, 0.0f);
                float iou = inter / (areaI + areaB - inter + 1e-9f);
                if (iou > NMS_TH) sms[i] = -__builtin_inff();
            }
            if (i == bi) sms[i] = -__builtin_inff();
        }
        __syncthreads();
    }

    // --- gather output [POSTN, 5] = (clipped box, sigmoid score) * valid
    for (int r = tid; r < POSTN; r += 1024) {
        int idx = selIdx[r];
        float v = selVal[r];
        float4 b = sbox[idx];
        float* o = out + ((size_t)n * POSTN + r) * 5;
        o[0] = b.x * v; o[1] = b.y * v; o[2] = b.z * v; o[3] = b.w * v;
        o[4] = ssig[idx] * v;
    }
}

// ------------------------------------------------------------- launcher ----
extern "C" void kernel_launch(void* const* d_in, const int* in_sizes, int n_in,
                              void* d_out, int out_size, void* d_ws, size_t ws_size,
                              hipStream_t stream) {
    (void)in_sizes; (void)n_in; (void)out_size; (void)ws_size;

    Ptrs p;
    for (int i = 0; i < NLVL; ++i) {           // dict order: obj0, box0, obj1, box1, ...
        p.obj[i] = (const float*)d_in[2 * i];
        p.box[i] = (const float*)d_in[2 * i + 1];
    }

    char* ws = (char*)d_ws;
    float*    candBox   = (float*)(ws + 0);          // N*5000*4 f32 = 160000 B
    float*    candScore = (float*)(ws + 160000);     // N*5000   f32 =  40000 B
    unsigned* hist      = (unsigned*)(ws + 200000);  // 10*256 u32   =  10240 B
    unsigned* prefix    = (unsigned*)(ws + 210240);  // 10 u32
    unsigned* krem      = (unsigned*)(ws + 210304);
    unsigned* cntgt     = (unsigned*)(ws + 210368);
    unsigned* cnteq     = (unsigned*)(ws + 210432);

    init_ws_kernel<<<157, 256, 0, stream>>>(candBox, candScore, hist, prefix, krem, cntgt, cnteq);

    dim3 grid(512, NIMG * NLVL);
    for (int pass = 0; pass < 4; ++pass) {
        hist_pass_kernel<<<grid, 256, 0, stream>>>(p, pass, hist, prefix);
        pick_digit_kernel<<<NIMG * NLVL, 1, 0, stream>>>(hist, prefix, krem);
    }
    compact_decode_kernel<<<grid, 256, 0, stream>>>(p, prefix, krem, cntgt, cnteq,
                                                    candBox, candScore);
    nms_kernel<<<NIMG, 1024, 0, stream>>>(candBox, candScore, (float*)d_out);
}